// GetVolume_71322226917788
// MI455X (gfx1250) — compile-verified
//
#include <hip/hip_runtime.h>
#include <hip/hip_bf16.h>

// Problem constants (match reference)
#define BB 2
#define CC 3
#define HH 256
#define WW 320
#define DD 64
#define HW (HH * WW)
#define INV_MIN 0.1f    // 1/MIN_DEPTH
#define INV_MAX 0.025f  // 1/MAX_DEPTH

// ---------------------------------------------------------------------------
// 4x4 inverse (adjugate) — runs on 4 threads total, cost is irrelevant.
// ---------------------------------------------------------------------------
__device__ inline void inv4x4(const float* m, float* o) {
    float i0  =  m[5]*m[10]*m[15] - m[5]*m[11]*m[14] - m[9]*m[6]*m[15] + m[9]*m[7]*m[14] + m[13]*m[6]*m[11] - m[13]*m[7]*m[10];
    float i4  = -m[4]*m[10]*m[15] + m[4]*m[11]*m[14] + m[8]*m[6]*m[15] - m[8]*m[7]*m[14] - m[12]*m[6]*m[11] + m[12]*m[7]*m[10];
    float i8  =  m[4]*m[9]*m[15]  - m[4]*m[11]*m[13] - m[8]*m[5]*m[15] + m[8]*m[7]*m[13] + m[12]*m[5]*m[11] - m[12]*m[7]*m[9];
    float i12 = -m[4]*m[9]*m[14]  + m[4]*m[10]*m[13] + m[8]*m[5]*m[14] - m[8]*m[6]*m[13] - m[12]*m[5]*m[10] + m[12]*m[6]*m[9];
    float i1  = -m[1]*m[10]*m[15] + m[1]*m[11]*m[14] + m[9]*m[2]*m[15] - m[9]*m[3]*m[14] - m[13]*m[2]*m[11] + m[13]*m[3]*m[10];
    float i5  =  m[0]*m[10]*m[15] - m[0]*m[11]*m[14] - m[8]*m[2]*m[15] + m[8]*m[3]*m[14] + m[12]*m[2]*m[11] - m[12]*m[3]*m[10];
    float i9  = -m[0]*m[9]*m[15]  + m[0]*m[11]*m[13] + m[8]*m[1]*m[15] - m[8]*m[3]*m[13] - m[12]*m[1]*m[11] + m[12]*m[3]*m[9];
    float i13 =  m[0]*m[9]*m[14]  - m[0]*m[10]*m[13] - m[8]*m[1]*m[14] + m[8]*m[2]*m[13] + m[12]*m[1]*m[10] - m[12]*m[2]*m[9];
    float i2  =  m[1]*m[6]*m[15]  - m[1]*m[7]*m[14]  - m[5]*m[2]*m[15] + m[5]*m[3]*m[14] + m[13]*m[2]*m[7]  - m[13]*m[3]*m[6];
    float i6  = -m[0]*m[6]*m[15]  + m[0]*m[7]*m[14]  + m[4]*m[2]*m[15] - m[4]*m[3]*m[14] - m[12]*m[2]*m[7]  + m[12]*m[3]*m[6];
    float i10 =  m[0]*m[5]*m[15]  - m[0]*m[7]*m[13]  - m[4]*m[1]*m[15] + m[4]*m[3]*m[13] + m[12]*m[1]*m[7]  - m[12]*m[3]*m[5];
    float i14 = -m[0]*m[5]*m[14]  + m[0]*m[6]*m[13]  + m[4]*m[1]*m[14] - m[4]*m[2]*m[13] - m[12]*m[1]*m[6]  + m[12]*m[2]*m[5];
    float i3  = -m[1]*m[6]*m[11]  + m[1]*m[7]*m[10]  + m[5]*m[2]*m[11] - m[5]*m[3]*m[10] - m[9]*m[2]*m[7]   + m[9]*m[3]*m[6];
    float i7  =  m[0]*m[6]*m[11]  - m[0]*m[7]*m[10]  - m[4]*m[2]*m[11] + m[4]*m[3]*m[10] + m[8]*m[2]*m[7]   - m[8]*m[3]*m[6];
    float i11 = -m[0]*m[5]*m[11]  + m[0]*m[7]*m[9]   + m[4]*m[1]*m[11] - m[4]*m[3]*m[9]  - m[8]*m[1]*m[7]   + m[8]*m[3]*m[5];
    float i15 =  m[0]*m[5]*m[10]  - m[0]*m[6]*m[9]   - m[4]*m[1]*m[10] + m[4]*m[2]*m[9]  + m[8]*m[1]*m[6]   - m[8]*m[2]*m[5];
    float det = m[0]*i0 + m[1]*i4 + m[2]*i8 + m[3]*i12;
    float r = 1.0f / det;
    o[0]=i0*r;  o[1]=i1*r;  o[2]=i2*r;  o[3]=i3*r;
    o[4]=i4*r;  o[5]=i5*r;  o[6]=i6*r;  o[7]=i7*r;
    o[8]=i8*r;  o[9]=i9*r;  o[10]=i10*r; o[11]=i11*r;
    o[12]=i12*r; o[13]=i13*r; o[14]=i14*r; o[15]=i15*r;
}

// Per (batch,image): M = src_proj @ inv(gt_proj); store rot(9)+trans(3).
// ws layout: [b][img][12] floats.
__global__ void proj_precompute(const float* __restrict__ lp,
                                const float* __restrict__ rp,
                                const float* __restrict__ gp,
                                float* __restrict__ ws) {
    int t = threadIdx.x;
    if (t >= BB * 2) return;
    int b = t >> 1, img = t & 1;
    const float* src = (img == 0 ? lp : rp) + b * 16;
    const float* g   = gp + b * 16;
    float inv[16];
    inv4x4(g, inv);
    float M[12];
    for (int i = 0; i < 3; ++i)
        for (int j = 0; j < 4; ++j) {
            float s = 0.f;
            for (int k = 0; k < 4; ++k) s = fmaf(src[i*4+k], inv[k*4+j], s);
            M[i*4+j] = s;
        }
    float* o = ws + (size_t)t * 12;
    o[0]=M[0]; o[1]=M[1]; o[2]=M[2];      // rot row 0
    o[3]=M[4]; o[4]=M[5]; o[5]=M[6];      // rot row 1
    o[6]=M[8]; o[7]=M[9]; o[8]=M[10];     // rot row 2
    o[9]=M[3]; o[10]=M[7]; o[11]=M[11];   // trans
}

// ---------------------------------------------------------------------------
// CDNA5 async global->LDS staging (ASYNCcnt path), inline asm (portable
// across ROCm7.2 / amdgpu-toolchain builtin-arity differences).
// GVS mode: mem = SGPR64 base + VGPR i32 byte offset; VDST = LDS byte addr.
// ---------------------------------------------------------------------------
__device__ __forceinline__ void async_load_f32_to_lds(unsigned lds_addr,
                                                      unsigned byte_off,
                                                      const float* base) {
    asm volatile("global_load_async_to_lds_b32 %0, %1, %2"
                 :: "v"(lds_addr), "v"(byte_off), "s"(base)
                 : "memory");
}
__device__ __forceinline__ void wait_async_le1() {
    asm volatile("s_wait_asynccnt 0x1" ::: "memory");
}
__device__ __forceinline__ void wait_async_0() {
    asm volatile("s_wait_asynccnt 0x0" ::: "memory");
}

// Bilinear sample of 3 channels, zero padding, align_corners semantics.
__device__ __forceinline__ void bilerp3(const float* __restrict__ img,
                                        float gx, float gy,
                                        float& s0, float& s1, float& s2) {
    // Pre-clamp so float->int cvt is in range; preserves validity decisions.
    gx = fminf(fmaxf(gx, -4.0f), (float)(WW + 4));
    gy = fminf(fmaxf(gy, -4.0f), (float)(HH + 4));
    float x0f = floorf(gx), y0f = floorf(gy);
    float wx1 = gx - x0f, wy1 = gy - y0f;
    float wx0 = 1.0f - wx1, wy0 = 1.0f - wy1;
    int x0 = (int)x0f, y0 = (int)y0f;
    int x1 = x0 + 1,  y1 = y0 + 1;
    bool vx0 = (unsigned)x0 < (unsigned)WW;
    bool vx1 = (unsigned)x1 < (unsigned)WW;
    bool vy0 = (unsigned)y0 < (unsigned)HH;
    bool vy1 = (unsigned)y1 < (unsigned)HH;
    int cx0 = min(max(x0, 0), WW - 1), cx1 = min(max(x1, 0), WW - 1);
    int cy0 = min(max(y0, 0), HH - 1), cy1 = min(max(y1, 0), HH - 1);
    float w00 = (vx0 && vy0) ? wx0 * wy0 : 0.0f;
    float w10 = (vx1 && vy0) ? wx1 * wy0 : 0.0f;
    float w01 = (vx0 && vy1) ? wx0 * wy1 : 0.0f;
    float w11 = (vx1 && vy1) ? wx1 * wy1 : 0.0f;
    int i00 = cy0 * WW + cx0, i10 = cy0 * WW + cx1;
    int i01 = cy1 * WW + cx0, i11 = cy1 * WW + cx1;
    const float* c0 = img;
    const float* c1 = img + HW;
    const float* c2 = img + 2 * HW;
    s0 = fmaf(c0[i00], w00, fmaf(c0[i10], w10, fmaf(c0[i01], w01, c0[i11] * w11)));
    s1 = fmaf(c1[i00], w00, fmaf(c1[i10], w10, fmaf(c1[i01], w01, c1[i11] * w11)));
    s2 = fmaf(c2[i00], w00, fmaf(c2[i10], w10, fmaf(c2[i01], w01, c2[i11] * w11)));
}

// One thread per pixel, loop over D. Tile = 64x4 (exact: 5*64=320, 64*4=256).
// depth_noise slices double-buffered through LDS via async loads.
__launch_bounds__(256)
__global__ void warp_volume(const float* __restrict__ limg,
                            const float* __restrict__ rimg,
                            const float* __restrict__ dn,
                            const float* __restrict__ ws,
                            float* __restrict__ out) {
    __shared__ float dbuf[512];  // 2 x 256 floats double buffer

    const int tid = threadIdx.x;
    const int tx = tid & 63, ty = tid >> 6;
    const int x = blockIdx.x * 64 + tx;
    const int y = blockIdx.y * 4 + ty;
    const int b = blockIdx.z;
    const float fx = (float)x, fy = (float)y;

    // Per-batch warp matrices (uniform across block -> scalar loads)
    const float* Ml = ws + (size_t)b * 24;
    const float* Mr = Ml + 12;
    const float rl0 = fmaf(Ml[0], fx, fmaf(Ml[1], fy, Ml[2]));
    const float rl1 = fmaf(Ml[3], fx, fmaf(Ml[4], fy, Ml[5]));
    const float rl2 = fmaf(Ml[6], fx, fmaf(Ml[7], fy, Ml[8]));
    const float tl0 = Ml[9], tl1 = Ml[10], tl2 = Ml[11];
    const float rr0 = fmaf(Mr[0], fx, fmaf(Mr[1], fy, Mr[2]));
    const float rr1 = fmaf(Mr[3], fx, fmaf(Mr[4], fy, Mr[5]));
    const float rr2 = fmaf(Mr[6], fx, fmaf(Mr[7], fy, Mr[8]));
    const float tr0 = Mr[9], tr1 = Mr[10], tr2 = Mr[11];

    const float* L = limg + (size_t)b * CC * HW;
    const float* R = rimg + (size_t)b * CC * HW;

    const unsigned pix = (unsigned)(y * WW + x);
    const unsigned dn_off0 = ((unsigned)b * DD * HW + pix) * 4u;
    const unsigned dstep = (unsigned)HW * 4u;
    const unsigned lds0 = (unsigned)(unsigned long long)&dbuf[tid];
    const unsigned lds1 = (unsigned)(unsigned long long)&dbuf[tid + 256];

    float* outp = out + (size_t)b * DD * HW + pix;

    // Prologue: stage depth_noise slice d=0
    async_load_f32_to_lds(lds0, dn_off0, dn);

    #pragma unroll 2
    for (int d = 0; d < DD; ++d) {
        const int cur = d & 1;
        if (d + 1 < DD) {
            async_load_f32_to_lds(cur ? lds0 : lds1,
                                  dn_off0 + (unsigned)(d + 1) * dstep, dn);
            wait_async_le1();   // slice d landed (in-order completion)
        } else {
            wait_async_0();
        }
        const float noise = dbuf[cur ? tid + 256 : tid];

        const float dsv = (noise + (float)d) * (1.0f / (float)DD);
        const float depth =
            __builtin_amdgcn_rcpf(fmaf(dsv, (INV_MIN - INV_MAX), INV_MAX));

        // Left projection
        float pzl = fmaf(rl2, depth, tl2);
        float pxl = fmaf(rl0, depth, tl0);
        float pyl = fmaf(rl1, depth, tl1);
        bool negl = pzl <= 0.001f;
        pxl = negl ? (float)WW : pxl;
        pyl = negl ? (float)HH : pyl;
        pzl = negl ? 1.0f : pzl;
        const float izl = __builtin_amdgcn_rcpf(pzl);
        float l0, l1, l2;
        bilerp3(L, pxl * izl, pyl * izl, l0, l1, l2);

        // Right projection
        float pzr = fmaf(rr2, depth, tr2);
        float pxr = fmaf(rr0, depth, tr0);
        float pyr = fmaf(rr1, depth, tr1);
        bool negr = pzr <= 0.001f;
        pxr = negr ? (float)WW : pxr;
        pyr = negr ? (float)HH : pyr;
        pzr = negr ? 1.0f : pzr;
        const float izr = __builtin_amdgcn_rcpf(pzr);
        float r0, r1, r2;
        bilerp3(R, pxr * izr, pyr * izr, r0, r1, r2);

        const float acc =
            fmaf(l0, r0, fmaf(l1, r1, l2 * r2)) * (1.0f / 3.0f);
        __builtin_nontemporal_store(acc, outp + (size_t)d * HW);
    }
}

extern "C" void kernel_launch(void* const* d_in, const int* in_sizes, int n_in,
                              void* d_out, int out_size, void* d_ws, size_t ws_size,
                              hipStream_t stream) {
    const float* limg  = (const float*)d_in[0];
    const float* rimg  = (const float*)d_in[1];
    const float* lproj = (const float*)d_in[2];
    const float* rproj = (const float*)d_in[3];
    const float* gproj = (const float*)d_in[4];
    const float* dnois = (const float*)d_in[5];
    float* ws = (float*)d_ws;

    proj_precompute<<<1, 32, 0, stream>>>(lproj, rproj, gproj, ws);

    dim3 grid(WW / 64, HH / 4, BB);   // 5 x 64 x 2
    warp_volume<<<grid, 256, 0, stream>>>(limg, rimg, dnois, ws, (float*)d_out);
}